// LocalMonopolyMoE_72722386256052
// MI455X (gfx1250) — compile-verified
//
#include <hip/hip_runtime.h>
#include <hip/hip_bf16.h>

// LocalMonopolyMoE: 8-expert dense VAE, per-sample argmin expert selection.
// Compute-bound (enc1/dec3 at ~430 FLOP/byte) -> fp32 WMMA (v_wmma_f32_16x16x4_f32)
// with Tensor Data Mover (TDM) double-buffered LDS tile staging.

#define E_      8
#define B_      1024
#define DFLAT_  8192
#define H_      512
#define Z_      128
#define NT_     (DFLAT_ / 64)   // 128 column tiles in dec3

typedef __attribute__((ext_vector_type(2))) float v2f;
typedef __attribute__((ext_vector_type(8))) float v8f;
typedef __attribute__((ext_vector_type(4))) unsigned int u32x4;
typedef __attribute__((ext_vector_type(8))) int i32x8;
typedef __attribute__((ext_vector_type(4))) int i32x4;

#if __has_builtin(__builtin_amdgcn_tensor_load_to_lds) && \
    __has_builtin(__builtin_amdgcn_s_wait_tensorcnt)
#define USE_TDM 1
#else
#define USE_TDM 0
#endif

#if USE_TDM
// ---------------------------------------------------------------------------
// Issue one 2D TDM tile load (global -> LDS), 4-byte elements.
// D# packing per CDNA5 ISA 8.3/8.4. padIntCode: 0=2,1=4,2=8,3=16,4=32,5=64
// DWORDs between pads; padAmtCode: pad of (code+1) DWORDs. padAmtCode<0 => no pad.
// Toolchain here exposes the 6-arg builtin:
//   (u32x4 group0, i32x8 group1, i32x4 group2, i32x4 group3, i32x8 extra, i32 cpol)
// ---------------------------------------------------------------------------
__device__ __forceinline__ void tdm_tile_2d(const float* gtile, unsigned lds_byte,
                                            unsigned tileX, unsigned tileY,
                                            unsigned tdim0, unsigned tdim1,
                                            unsigned long long stride0,
                                            int padIntCode, int padAmtCode)
{
    unsigned long long ga = (unsigned long long)gtile;
    u32x4 g0;
    g0[0] = 1u;                                             // count=1 (valid), flags=0
    g0[1] = lds_byte;                                       // lds_addr
    g0[2] = (unsigned)(ga & 0xFFFFFFFFu);                   // global_addr[31:0]
    g0[3] = (unsigned)((ga >> 32) & 0x01FFFFFFull) | (2u << 30); // addr[56:32], type=2

    unsigned w0 = (2u << 16);                               // data_size = 4 bytes
    if (padAmtCode >= 0)
        w0 |= (1u << 20) | ((unsigned)padIntCode << 22) | ((unsigned)padAmtCode << 25);
    i32x8 g1;
    g1[0] = (int)w0;                                        // mask=0, size, pad cfg
    g1[1] = (int)((tdim0 & 0xFFFFu) << 16);                 // tensor_dim0[15:0]
    g1[2] = (int)(((tdim0 >> 16) & 0xFFFFu) | ((tdim1 & 0xFFFFu) << 16));
    g1[3] = (int)(((tdim1 >> 16) & 0xFFFFu) | (tileX << 16));   // tile_dim0
    g1[4] = (int)(tileY & 0xFFFFu);                         // tile_dim1; tile_dim2=0
    g1[5] = (int)(stride0 & 0xFFFFFFFFull);                 // tensor_dim0_stride[31:0]
    g1[6] = (int)((stride0 >> 32) & 0xFFFFull);             // stride[47:32]; dim1_stride=0
    g1[7] = 0;
    i32x4 z4 = {0, 0, 0, 0};                                // groups 2/3 unused (2D)
    i32x8 z8 = {0, 0, 0, 0, 0, 0, 0, 0};
    __builtin_amdgcn_tensor_load_to_lds(g0, g1, z4, z4, z8, 0);
}
__device__ __forceinline__ unsigned lds_off(const void* p) {
    // generic LDS pointer: aperture in addr[63:32], LDS byte offset in addr[31:0]
    return (unsigned)(unsigned long long)(uintptr_t)p;
}
#endif

// ---------------------------------------------------------------------------
// Shared 64x64 WMMA tile pipeline (4 waves / 128 threads, K-step 32).
// A: [M,K] row-major, W: [K,N] row-major. Accumulates into acc0..acc3.
// ---------------------------------------------------------------------------
#define GEMM_CORE_BODY(Ae, We, M, N, K)                                         \
    __shared__ float As[2][64][36];                                             \
    __shared__ float Bs[2][32][68];                                             \
    const int tid  = threadIdx.x;                                               \
    const int lane = tid & 31;                                                  \
    const int wv   = tid >> 5;                                                  \
    const int mrow = lane & 15;                                                 \
    const int kh   = (lane >> 4) << 1;                                          \
    v8f acc0 = {}, acc1 = {}, acc2 = {}, acc3 = {};                             \
    const int NTILES = (K) / 32;                                                \
    TDM_PRELOAD(Ae, We, M, N, K)                                                \
    for (int t = 0; t < NTILES; ++t) {                                          \
        const int cur = t & 1;                                                  \
        STAGE_TILE(Ae, We, M, N, K)                                             \
        __syncthreads();                                                        \
        _Pragma("unroll")                                                       \
        for (int k0 = 0; k0 < 32; k0 += 4) {                                    \
            v2f a = *(const v2f*)&As[cur][wv * 16 + mrow][k0 + kh];             \
            v2f b0, b1, b2, b3;                                                 \
            b0.x = Bs[cur][k0 + kh][ 0 + mrow]; b0.y = Bs[cur][k0 + kh + 1][ 0 + mrow]; \
            b1.x = Bs[cur][k0 + kh][16 + mrow]; b1.y = Bs[cur][k0 + kh + 1][16 + mrow]; \
            b2.x = Bs[cur][k0 + kh][32 + mrow]; b2.y = Bs[cur][k0 + kh + 1][32 + mrow]; \
            b3.x = Bs[cur][k0 + kh][48 + mrow]; b3.y = Bs[cur][k0 + kh + 1][48 + mrow]; \
            acc0 = __builtin_amdgcn_wmma_f32_16x16x4_f32(false, a, false, b0, (short)0, acc0, false, false); \
            acc1 = __builtin_amdgcn_wmma_f32_16x16x4_f32(false, a, false, b1, (short)0, acc1, false, false); \
            acc2 = __builtin_amdgcn_wmma_f32_16x16x4_f32(false, a, false, b2, (short)0, acc2, false, false); \
            acc3 = __builtin_amdgcn_wmma_f32_16x16x4_f32(false, a, false, b3, (short)0, acc3, false, false); \
        }                                                                       \
        __syncthreads();                                                        \
    }

#if USE_TDM
// wave 0 drives the TDM: prefetch depth 1, ping-pong buffers.
#define TDM_ISSUE(Ae, We, M, N, K, kb, buf)                                     \
    tdm_tile_2d((Ae) + (long long)m0 * (K) + (kb), lds_off(&As[buf][0][0]),     \
                32u, 64u, (unsigned)(K), (unsigned)(M),                         \
                (unsigned long long)(unsigned)(K), 4, 3);                       \
    tdm_tile_2d((We) + (long long)(kb) * (N) + n0, lds_off(&Bs[buf][0][0]),     \
                64u, 32u, (unsigned)(N), (unsigned)(K),                         \
                (unsigned long long)(unsigned)(N), 5, 3);
#define TDM_PRELOAD(Ae, We, M, N, K)                                            \
    if (wv == 0) { TDM_ISSUE(Ae, We, M, N, K, 0, 0) }
#define STAGE_TILE(Ae, We, M, N, K)                                             \
    if (wv == 0) {                                                              \
        if (t + 1 < NTILES) {                                                   \
            TDM_ISSUE(Ae, We, M, N, K, (t + 1) * 32, (t + 1) & 1)               \
            __builtin_amdgcn_s_wait_tensorcnt((short)2);                        \
        } else {                                                                \
            __builtin_amdgcn_s_wait_tensorcnt((short)0);                        \
        }                                                                       \
    }
#else
// fallback: explicit float4 staging by all 128 threads (round-1 path)
#define TDM_PRELOAD(Ae, We, M, N, K)
#define STAGE_TILE(Ae, We, M, N, K)                                             \
    {                                                                           \
        const int kb = t * 32;                                                  \
        _Pragma("unroll")                                                       \
        for (int i = 0; i < 4; ++i) {                                           \
            int li = tid + i * 128;                                             \
            int r  = li >> 3, kq = (li & 7) << 2;                               \
            float4 v = *(const float4*)((Ae) + (long long)(m0 + r) * (K) + kb + kq); \
            *(float4*)(&As[cur][r][kq]) = v;                                    \
        }                                                                       \
        _Pragma("unroll")                                                       \
        for (int i = 0; i < 4; ++i) {                                           \
            int li = tid + i * 128;                                             \
            int r  = li >> 4, nq = (li & 15) << 2;                              \
            float4 v = *(const float4*)((We) + (long long)(kb + r) * (N) + n0 + nq); \
            *(float4*)(&Bs[cur][r][nq]) = v;                                    \
        }                                                                       \
    }
#endif

// ---------------------------------------------------------------------------
// Generic batched GEMM: C[e] = act(A[e] * W[e] + bias[e])
// ---------------------------------------------------------------------------
template<bool RELU>
__global__ __launch_bounds__(128)
void gemm_wmma(const float* __restrict__ A, long long sAe,
               const float* __restrict__ W, long long sWe,
               const float* __restrict__ bias, long long sBe,
               float* __restrict__ C, long long sCe,
               int M, int N, int K)
{
    const int e  = blockIdx.z;
    const float* Ae = A + (long long)e * sAe;
    const float* We = W + (long long)e * sWe;
    const float* be = bias + (long long)e * sBe;
    float*       Ce = C + (long long)e * sCe;
    const int m0 = blockIdx.y * 64;
    const int n0 = blockIdx.x * 64;

    GEMM_CORE_BODY(Ae, We, M, N, K)

    // C/D layout: VGPR j -> lanes 0-15: M=j, N=lane; lanes 16-31: M=8+j, N=lane-16
    const int rAdd = (lane >> 4) << 3;
    #pragma unroll
    for (int j = 0; j < 8; ++j) {
        int r = m0 + wv * 16 + rAdd + j;
        float v0 = acc0[j] + be[n0 +  0 + mrow];
        float v1 = acc1[j] + be[n0 + 16 + mrow];
        float v2 = acc2[j] + be[n0 + 32 + mrow];
        float v3 = acc3[j] + be[n0 + 48 + mrow];
        if (RELU) {
            v0 = fmaxf(v0, 0.f); v1 = fmaxf(v1, 0.f);
            v2 = fmaxf(v2, 0.f); v3 = fmaxf(v3, 0.f);
        }
        long long base = (long long)r * N + n0;
        Ce[base +  0 + mrow] = v0;
        Ce[base + 16 + mrow] = v1;
        Ce[base + 32 + mrow] = v2;
        Ce[base + 48 + mrow] = v3;
    }
}

// ---------------------------------------------------------------------------
// dec3 + loss (pass 1): xhat tile in-register, squared error vs xf reduced per
// row; deterministic per-column-tile partials (no float atomics).
// ---------------------------------------------------------------------------
__global__ __launch_bounds__(128)
void dec3_loss_wmma(const float* __restrict__ A, const float* __restrict__ W,
                    const float* __restrict__ bias, const float* __restrict__ xf,
                    float* __restrict__ part /* [E*B, NT_] */)
{
    const int e  = blockIdx.z;
    const float* Ae = A + (long long)e * B_ * H_;
    const float* We = W + (long long)e * H_ * DFLAT_;
    const float* be = bias + (long long)e * DFLAT_;
    const int m0 = blockIdx.y * 64;
    const int n0 = blockIdx.x * 64;
    const int M = B_, N = DFLAT_, K = H_;
    (void)M;

    GEMM_CORE_BODY(Ae, We, M, N, K)

    const int rAdd = (lane >> 4) << 3;
    #pragma unroll
    for (int j = 0; j < 8; ++j) {
        int r = m0 + wv * 16 + rAdd + j;
        const float* xr = xf + (long long)r * DFLAT_;
        float s = 0.f;
        { float v = acc0[j] + be[n0 +  0 + mrow]; float d = v - xr[n0 +  0 + mrow]; s += d * d; }
        { float v = acc1[j] + be[n0 + 16 + mrow]; float d = v - xr[n0 + 16 + mrow]; s += d * d; }
        { float v = acc2[j] + be[n0 + 32 + mrow]; float d = v - xr[n0 + 32 + mrow]; s += d * d; }
        { float v = acc3[j] + be[n0 + 48 + mrow]; float d = v - xr[n0 + 48 + mrow]; s += d * d; }
        s += __shfl_xor(s, 1, 16);
        s += __shfl_xor(s, 2, 16);
        s += __shfl_xor(s, 4, 16);
        s += __shfl_xor(s, 8, 16);
        if (mrow == 0)
            part[((long long)e * B_ + r) * NT_ + blockIdx.x] = s;
    }
}

// ---------------------------------------------------------------------------
// Small helper kernels
// ---------------------------------------------------------------------------
__global__ void z_kernel(const float* __restrict__ mu, const float* __restrict__ lv,
                         const float* __restrict__ eps, float* __restrict__ z, int n)
{
    int i = blockIdx.x * blockDim.x + threadIdx.x;
    if (i < n) z[i] = mu[i] + __expf(0.5f * lv[i]) * eps[i];
}

__global__ void loss_reduce(const float* __restrict__ part, float* __restrict__ sse)
{
    int i = blockIdx.x * blockDim.x + threadIdx.x;   // over E*B
    if (i < E_ * B_) {
        float s = 0.f;
        const float* p = part + (long long)i * NT_;
        #pragma unroll 8
        for (int t = 0; t < NT_; ++t) s += p[t];     // fixed order -> deterministic
        sse[i] = s;
    }
}

__global__ void argmin_kernel(const float* __restrict__ sse, int* __restrict__ idx,
                              float* __restrict__ out_idx)
{
    int b = blockIdx.x * blockDim.x + threadIdx.x;
    if (b >= B_) return;
    float best = sse[b];
    int bi = 0;
    #pragma unroll
    for (int e = 1; e < E_; ++e) {
        float v = sse[(long long)e * B_ + b];
        if (v < best) { best = v; bi = e; }          // strict < : first-index tiebreak
    }
    idx[b] = bi;
    out_idx[b] = (float)bi;
}

__global__ void gather_muvar(const float* __restrict__ mu, const float* __restrict__ lv,
                             const int* __restrict__ idx,
                             float* __restrict__ out_mu, float* __restrict__ out_lv)
{
    int t = blockIdx.x * blockDim.x + threadIdx.x;   // over B*Z
    if (t >= B_ * Z_) return;
    int b = t >> 7, j = t & (Z_ - 1);
    int e = idx[b];
    long long src = ((long long)e * B_ + b) * Z_ + j;
    out_mu[t] = mu[src];
    out_lv[t] = lv[src];
}

// dec3 pass 2: recompute xhat only for the winning expert of each sample.
// dec_w3 (128 MB total) fits in the 192 MB L2 -> repeated reads are L2 hits.
__global__ __launch_bounds__(256)
void xhat_sel_kernel(const float* __restrict__ hd2, const float* __restrict__ w3,
                     const float* __restrict__ b3, const int* __restrict__ idx,
                     float* __restrict__ out)
{
    __shared__ float hrow[H_];
    const int b = blockIdx.x;
    const int e = idx[b];
    const int tid = threadIdx.x;
    for (int i = tid; i < H_; i += 256)
        hrow[i] = hd2[((long long)e * B_ + b) * H_ + i];
    __syncthreads();
    const float* W  = w3 + (long long)e * H_ * DFLAT_;
    const float* bb = b3 + (long long)e * DFLAT_;
    for (int d = tid; d < DFLAT_; d += 256) {
        float s = bb[d];
        #pragma unroll 4
        for (int k = 0; k < H_; ++k)
            s = fmaf(hrow[k], W[(long long)k * DFLAT_ + d], s);
        out[(long long)b * DFLAT_ + d] = s;
    }
}

// ---------------------------------------------------------------------------
// Launch
// ---------------------------------------------------------------------------
extern "C" void kernel_launch(void* const* d_in, const int* in_sizes, int n_in,
                              void* d_out, int out_size, void* d_ws, size_t ws_size,
                              hipStream_t stream)
{
    const float* x      = (const float*)d_in[0];   // [B, DFLAT]
    const float* eps    = (const float*)d_in[1];   // [E,B,Z]
    const float* enc_w1 = (const float*)d_in[2];
    const float* enc_b1 = (const float*)d_in[3];
    const float* enc_w2 = (const float*)d_in[4];
    const float* enc_b2 = (const float*)d_in[5];
    const float* mu_w   = (const float*)d_in[6];
    const float* mu_b   = (const float*)d_in[7];
    const float* lv_w   = (const float*)d_in[8];
    const float* lv_b   = (const float*)d_in[9];
    const float* dec_w1 = (const float*)d_in[10];
    const float* dec_b1 = (const float*)d_in[11];
    const float* dec_w2 = (const float*)d_in[12];
    const float* dec_b2 = (const float*)d_in[13];
    const float* dec_w3 = (const float*)d_in[14];
    const float* dec_b3 = (const float*)d_in[15];

    // workspace layout (floats)
    float* ws  = (float*)d_ws;
    float* h1  = ws;                                   // E*B*H
    float* h2  = h1  + (size_t)E_ * B_ * H_;           // E*B*H
    float* mu  = h2  + (size_t)E_ * B_ * H_;           // E*B*Z
    float* lv  = mu  + (size_t)E_ * B_ * Z_;           // E*B*Z
    float* zz  = lv  + (size_t)E_ * B_ * Z_;           // E*B*Z
    float* hd1 = zz  + (size_t)E_ * B_ * Z_;           // E*B*H
    float* hd2 = hd1 + (size_t)E_ * B_ * H_;           // E*B*H
    float* prt = hd2 + (size_t)E_ * B_ * H_;           // E*B*NT_
    float* sse = prt + (size_t)E_ * B_ * NT_;          // E*B
    int*   idx = (int*)(sse + (size_t)E_ * B_);        // B

    // output: mu_sel [B*Z] | logvar_sel [B*Z] | xhat_sel [B*DFLAT] | idx [B]
    float* out      = (float*)d_out;
    float* out_mu   = out;
    float* out_lv   = out + (size_t)B_ * Z_;
    float* out_xhat = out + (size_t)2 * B_ * Z_;
    float* out_idx  = out + (size_t)2 * B_ * Z_ + (size_t)B_ * DFLAT_;

    const dim3 blk(128);

    // enc1: [B,DFLAT] x [DFLAT,H] (+relu), x shared across experts
    gemm_wmma<true><<<dim3(H_ / 64, B_ / 64, E_), blk, 0, stream>>>(
        x, 0LL, enc_w1, (long long)DFLAT_ * H_, enc_b1, H_, h1, (long long)B_ * H_,
        B_, H_, DFLAT_);
    // enc2: [B,H] x [H,H] (+relu)
    gemm_wmma<true><<<dim3(H_ / 64, B_ / 64, E_), blk, 0, stream>>>(
        h1, (long long)B_ * H_, enc_w2, (long long)H_ * H_, enc_b2, H_, h2, (long long)B_ * H_,
        B_, H_, H_);
    // mu / logvar heads: [B,H] x [H,Z]
    gemm_wmma<false><<<dim3(Z_ / 64, B_ / 64, E_), blk, 0, stream>>>(
        h2, (long long)B_ * H_, mu_w, (long long)H_ * Z_, mu_b, Z_, mu, (long long)B_ * Z_,
        B_, Z_, H_);
    gemm_wmma<false><<<dim3(Z_ / 64, B_ / 64, E_), blk, 0, stream>>>(
        h2, (long long)B_ * H_, lv_w, (long long)H_ * Z_, lv_b, Z_, lv, (long long)B_ * Z_,
        B_, Z_, H_);
    // z = mu + exp(0.5*lv)*eps
    {
        int n = E_ * B_ * Z_;
        z_kernel<<<(n + 255) / 256, 256, 0, stream>>>(mu, lv, eps, zz, n);
    }
    // dec1: [B,Z] x [Z,H] (+relu)
    gemm_wmma<true><<<dim3(H_ / 64, B_ / 64, E_), blk, 0, stream>>>(
        zz, (long long)B_ * Z_, dec_w1, (long long)Z_ * H_, dec_b1, H_, hd1, (long long)B_ * H_,
        B_, H_, Z_);
    // dec2: [B,H] x [H,H] (+relu)
    gemm_wmma<true><<<dim3(H_ / 64, B_ / 64, E_), blk, 0, stream>>>(
        hd1, (long long)B_ * H_, dec_w2, (long long)H_ * H_, dec_b2, H_, hd2, (long long)B_ * H_,
        B_, H_, H_);
    // dec3 pass 1: fused xhat + per-row SSE partials (xhat never materialized)
    dec3_loss_wmma<<<dim3(DFLAT_ / 64, B_ / 64, E_), blk, 0, stream>>>(
        hd2, dec_w3, dec_b3, x, prt);
    loss_reduce<<<(E_ * B_ + 255) / 256, 256, 0, stream>>>(prt, sse);
    // argmin over experts (first-index tiebreak), gather mu/logvar
    argmin_kernel<<<(B_ + 255) / 256, 256, 0, stream>>>(sse, idx, out_idx);
    gather_muvar<<<(B_ * Z_ + 255) / 256, 256, 0, stream>>>(mu, lv, idx, out_mu, out_lv);
    // dec3 pass 2: recompute xhat for winning expert only
    xhat_sel_kernel<<<dim3(B_), dim3(256), 0, stream>>>(hd2, dec_w3, dec_b3, idx, out_xhat);
}